// KNN_52828097740973
// MI455X (gfx1250) — compile-verified
//
#include <hip/hip_runtime.h>

#define NF 128   // feature dim
#define NK 64    // neighbors
#define SENT 0x7f800001u

typedef __attribute__((ext_vector_type(2))) float v2f;
typedef __attribute__((ext_vector_type(8))) float v8f;
typedef int v4i_gnu __attribute__((vector_size(16)));   // B128 payload type per clang sig

__device__ __forceinline__ float finf() { return __builtin_inff(); }

// ---- gfx1250 async memory->LDS copy (ASYNCcnt-tracked, no VGPR data) ----
__device__ __forceinline__ void async_cp16(const float4* g, float4* l) {
#if __has_builtin(__builtin_amdgcn_global_load_async_to_lds_b128)
  __builtin_amdgcn_global_load_async_to_lds_b128(
      (__attribute__((address_space(1))) v4i_gnu*)g,
      (__attribute__((address_space(3))) v4i_gnu*)l, 0, 0);
#else
  unsigned lofs = (unsigned)(size_t)l;   // low 32 bits of flat LDS addr = LDS offset
  asm volatile("global_load_async_to_lds_b128 %0, %1, off"
               :: "v"(lofs), "v"(g) : "memory");
#endif
}

__device__ __forceinline__ void wait_async0() {
#if __has_builtin(__builtin_amdgcn_s_wait_asynccnt)
  __builtin_amdgcn_s_wait_asynccnt(0);
#else
  asm volatile("s_wait_asynccnt 0x0" ::: "memory");
#endif
}

// ---------------- Kernel 0: row squared norms ----------------
__global__ void sq_kernel(const float* __restrict__ emb, float* __restrict__ g_sq, int N) {
  int i = blockIdx.x * blockDim.x + threadIdx.x;
  if (i >= N) return;
  const float4* p = (const float4*)(emb + (size_t)i * NF);
  float s = 0.f;
#pragma unroll
  for (int t = 0; t < NF / 4; ++t) {
    float4 f = p[t];
    s += f.x * f.x + f.y * f.y + f.z * f.z + f.w * f.w;
  }
  g_sq[i] = s;
}

// Issue async staging of 128 columns (= 128 contiguous emb rows) into sB.
// Columns past N are zero-padded with regular ds_stores (disjoint addresses).
__device__ __forceinline__ void stage_async(const float* __restrict__ emb, int col0, int N,
                                            float* sB, int tid) {
  const float4* src = (const float4*)(emb + (size_t)col0 * NF);
  float4* dst = (float4*)sB;
  int nv = N - col0; if (nv > 128) nv = 128;
  const int nv4 = nv * (NF / 4);
  for (int e = tid; e < 128 * NF / 4; e += 256) {
    if (e < nv4) async_cp16(src + e, dst + e);
    else dst[e] = make_float4(0.f, 0.f, 0.f, 0.f);
  }
}

// ---------------- Kernel 1: fused WMMA distance + top-64 selection ----------------
// One workgroup = 16 rows, 8 waves. Double-buffered async staging of 128 columns;
// each wave computes one 16x16 G-tile with 32x v_wmma_f32_16x16x4_f32 (K=128),
// forms d2, then merges 128 candidates/row into a sorted ascending top-64 list.
__launch_bounds__(256)
__global__ void knn_kernel(const float* __restrict__ emb, const float* __restrict__ g_sq,
                           int* __restrict__ g_idx, int N) {
  extern __shared__ char smem[];
  float* sA  = (float*)smem;              // 16 x 128 f32   (8 KB)
  float* sB0 = sA + 16 * NF;              // 128 x 128 f32  (64 KB)
  float* sB1 = sB0 + 128 * NF;            // 128 x 128 f32  (64 KB)
  float* sD  = sB1 + 128 * NF;            // 16 x 128 f32   (8 KB)
  float* sLd = sD + 16 * 128;             // 16 x 64 dists  (4 KB)
  int*   sLi = (int*)(sLd + 16 * NK);     // 16 x 64 idx    (4 KB)
  float* sSq = (float*)(sLi + 16 * NK);   // 16 row norms

  const int tid  = threadIdx.x;
  const int wave = tid >> 5;
  const int lane = tid & 31;
  const int half = lane >> 4;   // 0: lanes 0-15, 1: lanes 16-31
  const int lr   = lane & 15;
  const int row0 = blockIdx.x * 16;

  // Kick off async staging of round 0 while we set up A tile and lists.
  stage_async(emb, 0, N, sB0, tid);

  // Load A tile (16 contiguous rows of emb), zero-pad past N.
  {
    const float4* src = (const float4*)(emb + (size_t)row0 * NF);
    float4* dst = (float4*)sA;
    int nrow = N - row0; if (nrow > 16) nrow = 16;
    const int nv4 = nrow * (NF / 4);
    for (int e = tid; e < 16 * NF / 4; e += 256)
      dst[e] = (e < nv4) ? src[e] : make_float4(0.f, 0.f, 0.f, 0.f);
  }
  if (tid < 16) sSq[tid] = (row0 + tid < N) ? g_sq[row0 + tid] : 0.f;
  for (int e = tid; e < 16 * NK; e += 256) { sLd[e] = finf(); sLi[e] = 0; }
  wait_async0();
  __syncthreads();

  const int rounds = (N + 127) >> 7;
  for (int rnd = 0; rnd < rounds; ++rnd) {
    const int col0 = rnd << 7;
    float* sB     = (rnd & 1) ? sB1 : sB0;
    float* sBnext = (rnd & 1) ? sB0 : sB1;

    // Prefetch next round's columns asynchronously (buffer free since rnd-1).
    if (rnd + 1 < rounds) stage_async(emb, (rnd + 1) << 7, N, sBnext, tid);

    // GEMM tile: rows [row0,row0+16) x cols [col0+wave*16, +16), exact f32 WMMA.
    // A 16x4 frag: lanes 0-15 K={4kk,4kk+1}, lanes 16-31 K={4kk+2,4kk+3}; B symmetric.
    v8f acc = {};
    {
      const float* pa = sA + lr * NF;
      const float* pb = sB + (wave * 16 + lr) * NF;
      const int koff = half * 2;
#pragma unroll
      for (int kk = 0; kk < NF / 4; ++kk) {
        v2f a = *(const v2f*)(pa + kk * 4 + koff);
        v2f b = *(const v2f*)(pb + kk * 4 + koff);
        acc = __builtin_amdgcn_wmma_f32_16x16x4_f32(false, a, false, b,
                                                    (short)0, acc, false, false);
      }
    }
    // d2 = sq_i + sq_j - 2*G; diagonal and padded columns -> +inf.
    {
      const int jg = col0 + wave * 16 + lr;
      const float sqj = (jg < N) ? g_sq[jg] : 0.f;
#pragma unroll
      for (int v = 0; v < 8; ++v) {
        const int m = v + half * 8;                 // C/D layout: VGPR v -> M
        float d2 = sSq[m] + sqj - 2.f * acc[v];
        if (jg >= N || jg == row0 + m) d2 = finf();
        sD[m * 128 + wave * 16 + lr] = d2;
      }
    }
    __syncthreads();

    // Selection: wave owns rows 2*wave, 2*wave+1; merge 128 candidates each.
#pragma unroll
    for (int rr = 0; rr < 2; ++rr) {
      const int m = wave * 2 + rr;
      float* ld = sLd + m * NK;
      int*   li = sLi + m * NK;
      for (int t = 0; t < 4; ++t) {
        const int c = lane + t * 32;
        const float d = sD[m * 128 + c];
        const int j = col0 + c;
        unsigned long long mask = __ballot(d < ld[NK - 1]);
        while (mask) {                              // uniform across wave
          const int l = __builtin_ctzll(mask);      // ascending lane = ascending j
          mask &= mask - 1;
          const float dl = __shfl(d, l);
          const int   jl = __shfl(j, l);
          if (dl < ld[NK - 1]) {                    // uniform re-check
            int lo = 0, hi = NK;                    // upper_bound -> stable ties
            while (lo < hi) { int mid = (lo + hi) >> 1;
                              if (ld[mid] <= dl) lo = mid + 1; else hi = mid; }
            const int p = lo;
            const int q0 = p + 1 + lane, q1 = q0 + 32;
            float v0 = 0.f, v1 = 0.f; int i0 = 0, i1 = 0;
            if (q0 < NK) { v0 = ld[q0 - 1]; i0 = li[q0 - 1]; }
            if (q1 < NK) { v1 = ld[q1 - 1]; i1 = li[q1 - 1]; }
            __builtin_amdgcn_wave_barrier();        // reads before shift writes
            if (q0 < NK) { ld[q0] = v0; li[q0] = i0; }
            if (q1 < NK) { ld[q1] = v1; li[q1] = i1; }
            if (lane == 0) { ld[p] = dl; li[p] = jl; }
            __builtin_amdgcn_wave_barrier();
          }
        }
      }
    }
    // Next round's staged buffer must be fully landed before compute.
    wait_async0();
    __syncthreads();
  }

  // Emit sorted-ascending neighbor indices.
  for (int e = tid; e < 16 * NK; e += 256) {
    const int m = e >> 6, k = e & 63;
    if (row0 + m < N) g_idx[(size_t)(row0 + m) * NK + k] = sLi[e];
  }
}

// ---------------- Kernel 2: samples = [node | neighbor | weight] ----------------
__global__ void samples_kernel(const float* __restrict__ emb, const float* __restrict__ sw,
                               const int* __restrict__ g_idx, float* __restrict__ out, int N) {
  const int row = blockIdx.x;                 // [0, N*NK)
  const int i = row >> 6;
  const int j = g_idx[row];
  float* o = out + (size_t)row * (2 * NF + 1);
  const int c = threadIdx.x;                  // blockDim = 128
  o[c]      = emb[(size_t)i * NF + c];
  o[NF + c] = emb[(size_t)j * NF + c];
  if (c == 0) o[2 * NF] = sw[j];
}

// ---------------- Kernel 3a/3b/3c: adj = cumulative scatter ----------------
__global__ void fill_kernel(float4* __restrict__ adj4, long long n4,
                            float* __restrict__ tail, int tailn) {
  long long e = (long long)blockIdx.x * blockDim.x + threadIdx.x;
  const float s = __uint_as_float(SENT);
  if (e < n4) adj4[e] = make_float4(s, s, s, s);
  if (e == 0) for (int t = 0; t < tailn; ++t) tail[t] = s;
}

__global__ void scatter_kernel(const int* __restrict__ g_idx, const float* __restrict__ eo,
                               float* __restrict__ adj, int N) {
  int i = blockIdx.x * blockDim.x + threadIdx.x;
  if (i >= N) return;
  float* rowp = adj + (size_t)i * N;
  const int* ip = g_idx + (size_t)i * NK;
  const float* ep = eo + (size_t)i * NK;
  for (int k = 0; k < NK; ++k) rowp[ip[k]] = ep[k];   // serial k: last write wins
}

__global__ void scan_kernel(float* __restrict__ adj, int N) {
  int j = blockIdx.x * blockDim.x + threadIdx.x;      // column forward-fill
  if (j >= N) return;
  float carry = 0.f;
  for (int i = 0; i < N; ++i) {
    float* p = adj + (size_t)i * N + j;               // coalesced across lanes
    float v = *p;
    if (__float_as_uint(v) == SENT) *p = carry; else carry = v;
  }
}

// ---------------- Launch ----------------
extern "C" void kernel_launch(void* const* d_in, const int* in_sizes, int n_in,
                              void* d_out, int out_size, void* d_ws, size_t ws_size,
                              hipStream_t stream) {
  (void)n_in; (void)out_size; (void)ws_size;
  const float* emb = (const float*)d_in[0];   // [N, 128]
  const float* sw  = (const float*)d_in[1];   // [N]
  const float* eo  = (const float*)d_in[2];   // [N*64]
  float* out = (float*)d_out;
  const int N = in_sizes[1];

  float* g_sq = (float*)d_ws;
  int* g_idx  = (int*)((char*)d_ws + (((size_t)N * sizeof(float) + 255) & ~(size_t)255));

  sq_kernel<<<(N + 255) / 256, 256, 0, stream>>>(emb, g_sq, N);

  const int nblk = (N + 15) / 16;
  const size_t smem = (16 * NF + 2 * 128 * NF + 16 * 128 + 16 * NK) * sizeof(float)
                    + 16 * NK * sizeof(int) + 16 * sizeof(float);   // ~152 KB dynamic LDS
  knn_kernel<<<nblk, 256, smem, stream>>>(emb, g_sq, g_idx, N);

  samples_kernel<<<N * NK, 128, 0, stream>>>(emb, sw, g_idx, out, N);

  float* adj = out + (size_t)N * NK * (2 * NF + 1);
  long long total = (long long)N * N;
  long long n4 = total >> 2;
  fill_kernel<<<(unsigned)((n4 + 255) / 256), 256, 0, stream>>>(
      (float4*)adj, n4, adj + n4 * 4, (int)(total & 3));
  scatter_kernel<<<(N + 255) / 256, 256, 0, stream>>>(g_idx, eo, adj, N);
  scan_kernel<<<(N + 255) / 256, 256, 0, stream>>>(adj, N);
}